// Dilation_79018808311965
// MI455X (gfx1250) — compile-verified
//
#include <hip/hip_runtime.h>
#include <cstdint>

// Dilation with 5x5 all-ones SE == 5x5 sliding max, stride 1, same padding.
// Pad value -1e4 never wins (inputs are uniform[0,1) >= 0 and every window
// contains its in-bounds center), so border-replicated (index-clamped) max
// is exactly the reference output.

#define IMG_H 2048
#define IMG_W 2048
#define TILE  64
#define HALO  2
#define LROWS (TILE + 2 * HALO)   // 68
#define LCOLS (TILE + 2 * HALO)   // 68

typedef __attribute__((ext_vector_type(4))) unsigned int v4u;
typedef __attribute__((ext_vector_type(8))) int v8i;
typedef __attribute__((ext_vector_type(4))) int v4i;

__global__ __launch_bounds__(256) void Dilation_79018808311965_kernel(
    const float* __restrict__ img, float* __restrict__ out) {
  __shared__ float tile[LROWS * LCOLS];

  const int x0 = blockIdx.x * TILE;
  const int y0 = blockIdx.y * TILE;
  const long long plane_off =
      (long long)blockIdx.z * (long long)(IMG_H * IMG_W);

  // Clamp-shifted 68x68 source window: always fully in-bounds, fixed shape.
  int gx0 = x0 - HALO;
  if (gx0 < 0) gx0 = 0;
  if (gx0 > IMG_W - LCOLS) gx0 = IMG_W - LCOLS;
  int gy0 = y0 - HALO;
  if (gy0 < 0) gy0 = 0;
  if (gy0 > IMG_H - LROWS) gy0 = IMG_H - LROWS;

#if __has_builtin(__builtin_amdgcn_tensor_load_to_lds)
  // --- TDM path: one wave DMAs the 68x68 fp32 tile global -> LDS ---------
  if (threadIdx.x < 32) {  // wave 0 only (uniform branch; TDM ignores EXEC)
    const unsigned long long gaddr =
        (unsigned long long)(uintptr_t)img +
        4ull * (unsigned long long)(plane_off + (long long)gy0 * IMG_W + gx0);
    const unsigned lds_base = (unsigned)(uintptr_t)(&tile[0]);  // LDS byte offset

    v4u g0;
    g0.x = 1u;                                        // count=1, user load D#
    g0.y = lds_base;                                  // lds_addr
    g0.z = (unsigned)(gaddr & 0xFFFFFFFFull);         // global_addr[31:0]
    g0.w = (unsigned)((gaddr >> 32) & 0x1FFFFFFull)   // global_addr[56:32]
           | (2u << 30);                              // type = 2 ("image")

    const unsigned long long s1 =
        (unsigned long long)IMG_W * (unsigned long long)IMG_H;
    v8i g1;
    g1[0] = (int)(2u << 16);                          // data_size = 4B
    g1[1] = (int)((unsigned)(IMG_W & 0xFFFF) << 16);  // tensor_dim0[15:0]
    g1[2] = (int)((((unsigned)IMG_W >> 16) & 0xFFFFu) // tensor_dim0[31:16]
                  | ((unsigned)(IMG_H & 0xFFFF) << 16)); // tensor_dim1[15:0]
    g1[3] = (int)((((unsigned)IMG_H >> 16) & 0xFFFFu)
                  | ((unsigned)LCOLS << 16));         // tile_dim0 = 68
    g1[4] = (int)(unsigned)LROWS;                     // tile_dim1 = 68, tile_dim2 = 0
    g1[5] = (int)(unsigned)IMG_W;                     // tensor_dim0_stride[31:0]
    g1[6] = (int)((unsigned)(s1 & 0xFFFFull) << 16);  // stride0[47:32]=0 | stride1[15:0]
    g1[7] = (int)((s1 >> 16) & 0xFFFFFFFFull);        // tensor_dim1_stride[47:16]

    v4i g2 = {0, 0, 0, 0};
    v4i g3 = {0, 0, 0, 0};
#if defined(__clang_major__) && __clang_major__ >= 23
    v8i g4 = {0, 0, 0, 0, 0, 0, 0, 0};
    __builtin_amdgcn_tensor_load_to_lds(g0, g1, g2, g3, g4, 0);
#else
    __builtin_amdgcn_tensor_load_to_lds(g0, g1, g2, g3, 0);
#endif
    __builtin_amdgcn_s_wait_tensorcnt(0);
  }
#else
  // --- Fallback: cooperative vector-load path -----------------------------
  for (int i = threadIdx.x; i < LROWS * LCOLS; i += 256) {
    const int r = i / LCOLS, c = i - r * LCOLS;
    tile[i] = img[plane_off + (long long)(gy0 + r) * IMG_W + (gx0 + c)];
  }
#endif
  __syncthreads();

  // --- Separable 5x5 max from LDS; each thread emits a 4x4 patch ----------
  const int tx = threadIdx.x & 15;
  const int ty = threadIdx.x >> 4;
  const int ox = x0 + tx * 4;   // first output column
  const int oy = y0 + ty * 4;   // first output row

  // Clamped (border-replicated) LDS indices for cols ox-2..ox+5, rows oy-2..oy+5.
  int lcol[8], lrow[8];
#pragma unroll
  for (int j = 0; j < 8; ++j) {
    int gc = ox - HALO + j;
    gc = gc < 0 ? 0 : (gc > IMG_W - 1 ? IMG_W - 1 : gc);
    lcol[j] = gc - gx0;
    int gr = oy - HALO + j;
    gr = gr < 0 ? 0 : (gr > IMG_H - 1 ? IMG_H - 1 : gr);
    lrow[j] = gr - gy0;
  }

  // Horizontal 5-wide max for 8 rows x 4 output columns.
  float h[8][4];
#pragma unroll
  for (int r = 0; r < 8; ++r) {
    const float* rowp = &tile[lrow[r] * LCOLS];
    const float v0 = rowp[lcol[0]], v1 = rowp[lcol[1]];
    const float v2 = rowp[lcol[2]], v3 = rowp[lcol[3]];
    const float v4 = rowp[lcol[4]], v5 = rowp[lcol[5]];
    const float v6 = rowp[lcol[6]], v7 = rowp[lcol[7]];
    const float m12 = fmaxf(v1, v2), m34 = fmaxf(v3, v4), m56 = fmaxf(v5, v6);
    const float m1234 = fmaxf(m12, m34);          // max(v1..v4)
    const float m3456 = fmaxf(m34, m56);          // max(v3..v6)
    h[r][0] = fmaxf(v0, m1234);                   // max(v0..v4)
    h[r][1] = fmaxf(m1234, v5);                   // max(v1..v5)
    h[r][2] = fmaxf(v2, m3456);                   // max(v2..v6)
    h[r][3] = fmaxf(m3456, v7);                   // max(v3..v7)
  }

  // Vertical 5-tall max, coalesced float4 stores.
  float* op = out + plane_off + (long long)oy * IMG_W + ox;
#pragma unroll
  for (int r = 0; r < 4; ++r) {
    float4 o;
    o.x = fmaxf(fmaxf(fmaxf(h[r][0], h[r + 1][0]),
                      fmaxf(h[r + 2][0], h[r + 3][0])), h[r + 4][0]);
    o.y = fmaxf(fmaxf(fmaxf(h[r][1], h[r + 1][1]),
                      fmaxf(h[r + 2][1], h[r + 3][1])), h[r + 4][1]);
    o.z = fmaxf(fmaxf(fmaxf(h[r][2], h[r + 1][2]),
                      fmaxf(h[r + 2][2], h[r + 3][2])), h[r + 4][2]);
    o.w = fmaxf(fmaxf(fmaxf(h[r][3], h[r + 1][3]),
                      fmaxf(h[r + 2][3], h[r + 3][3])), h[r + 4][3]);
    *(float4*)(op + (long long)r * IMG_W) = o;
  }
}

extern "C" void kernel_launch(void* const* d_in, const int* in_sizes, int n_in,
                              void* d_out, int out_size, void* d_ws, size_t ws_size,
                              hipStream_t stream) {
  const float* img = (const float*)d_in[0];
  // d_in[1] is the 5x5 all-ones SE: additive term is all zeros -> pure max filter.
  float* out = (float*)d_out;

  const int planes = in_sizes[0] / (IMG_H * IMG_W);  // B*C = 24
  dim3 grid(IMG_W / TILE, IMG_H / TILE, planes);
  Dilation_79018808311965_kernel<<<grid, 256, 0, stream>>>(img, out);
}